// StructuralLoss_43739946942787
// MI455X (gfx1250) — compile-verified
//
#include <hip/hip_runtime.h>

typedef __attribute__((ext_vector_type(2))) float v2f;
typedef __attribute__((ext_vector_type(8))) float v8f;

#define TILE   16
#define HALO   5
#define LROWS  32      // 26 used rows + zeroed tail so WMMA never sees garbage
#define LCOLS  28      // 26 used cols padded to 7 K-chunks of 4
#define NCH    3
#define IMG    512
#define NTX    (IMG / TILE)          // 32 tiles per axis
#define NTILES (16 * NTX * NTX)      // 16384 workgroups (one wave each)

// band(i,k) == 1 iff input index k contributes to output index i of an
// 11-wide box: k in [i, i+10] (input staged with -5 halo offset).
__device__ __forceinline__ float band(int i, int k) {
    return (k >= i && k <= i + 10) ? 1.0f : 0.0f;
}

__global__ __launch_bounds__(32) void
ssim_tile_kernel(const float* __restrict__ pred,
                 const float* __restrict__ gt,
                 float* __restrict__ partial)
{
    __shared__ float Pl[LROWS * LCOLS];
    __shared__ float Gl[LROWS * LCOLS];
    __shared__ float Hl[5][LROWS * TILE];

    const int tid = threadIdx.x;      // 0..31, single wave32
    const int lid = tid & 15;
    const int hi  = tid >> 4;

    const int blk = blockIdx.x;
    const int b   = blk >> 10;        // / (32*32)
    const int rem = blk & 1023;
    const int y0  = (rem >> 5) * TILE;
    const int x0  = (rem & 31) * TILE;

    // Constant 0/1 band fragments. Identical per-lane values serve as the
    // B operand (horizontal pass) and the A operand (vertical pass):
    // both map lane%16 -> the "output" index and 2*(lane/16)+{0,1} -> K.
    v2f w[7];
#pragma unroll
    for (int t = 0; t < 7; ++t) {
        const int k0 = 4 * t + 2 * hi;
        w[t].x = band(lid, k0);
        w[t].y = band(lid, k0 + 1);
    }

    // Horizontal accumulators: 5 quantities x 2 row-blocks (rows 0..15, 16..31).
    // Channel sum folds in here via the WMMA C operand.
    v8f hc[5][2];
#pragma unroll
    for (int q = 0; q < 5; ++q) { hc[q][0] = (v8f){}; hc[q][1] = (v8f){}; }

    for (int c = 0; c < NCH; ++c) {
        const float* pbase = pred + (size_t)(b * NCH + c) * IMG * IMG;
        const float* gbase = gt   + (size_t)(b * NCH + c) * IMG * IMG;

        __syncthreads();
        // Stage 32x28 zero-padded halo tiles of p and g into LDS.
        for (int e = tid; e < LROWS * LCOLS; e += 32) {
            const int r  = e / LCOLS;
            const int cc = e - r * LCOLS;
            const int gy = y0 + r - HALO;
            const int gx = x0 + cc - HALO;
            float pv = 0.0f, gv = 0.0f;
            if (r < 26 && gy >= 0 && gy < IMG && gx >= 0 && gx < IMG) {
                const size_t o = (size_t)gy * IMG + gx;
                pv = pbase[o];
                gv = gbase[o];
            }
            Pl[e] = pv;
            Gl[e] = gv;
        }
        __syncthreads();

        // Horizontal pass: H = Q * Band.  A = data (16x4 chunks), B = band.
#pragma unroll
        for (int t = 0; t < 7; ++t) {
            const int cc0 = 4 * t + 2 * hi;
#pragma unroll
            for (int rb = 0; rb < 2; ++rb) {
                const int row = lid + 16 * rb;
                const float p0 = Pl[row * LCOLS + cc0];
                const float p1 = Pl[row * LCOLS + cc0 + 1];
                const float g0 = Gl[row * LCOLS + cc0];
                const float g1 = Gl[row * LCOLS + cc0 + 1];
                const v2f ap  = {p0, p1};
                const v2f ag  = {g0, g1};
                const v2f app = {p0 * p0, p1 * p1};
                const v2f agg = {g0 * g0, g1 * g1};
                const v2f apg = {p0 * g0, p1 * g1};
                hc[0][rb] = __builtin_amdgcn_wmma_f32_16x16x4_f32(false, ap,  false, w[t], (short)0, hc[0][rb], false, false);
                hc[1][rb] = __builtin_amdgcn_wmma_f32_16x16x4_f32(false, ag,  false, w[t], (short)0, hc[1][rb], false, false);
                hc[2][rb] = __builtin_amdgcn_wmma_f32_16x16x4_f32(false, app, false, w[t], (short)0, hc[2][rb], false, false);
                hc[3][rb] = __builtin_amdgcn_wmma_f32_16x16x4_f32(false, agg, false, w[t], (short)0, hc[3][rb], false, false);
                hc[4][rb] = __builtin_amdgcn_wmma_f32_16x16x4_f32(false, apg, false, w[t], (short)0, hc[4][rb], false, false);
            }
        }
    }

    // Re-shape H from C/D layout (M = vgpr + 8*hi + 16*rb, N = lid) into
    // plain row-major LDS so it can be re-read as a B operand.
    __syncthreads();
#pragma unroll
    for (int q = 0; q < 5; ++q)
#pragma unroll
        for (int rb = 0; rb < 2; ++rb)
#pragma unroll
            for (int v = 0; v < 8; ++v) {
                const int row = v + 8 * hi + 16 * rb;
                Hl[q][row * TILE + lid] = hc[q][rb][v];
            }
    __syncthreads();

    // Vertical pass: S = Band * H.  A = band fragments, B = H chunks.
    v8f s[5];
#pragma unroll
    for (int q = 0; q < 5; ++q) s[q] = (v8f){};
#pragma unroll
    for (int t = 0; t < 7; ++t) {
        const int r0 = 4 * t + 2 * hi;
#pragma unroll
        for (int q = 0; q < 5; ++q) {
            const v2f bq = {Hl[q][r0 * TILE + lid], Hl[q][(r0 + 1) * TILE + lid]};
            s[q] = __builtin_amdgcn_wmma_f32_16x16x4_f32(false, w[t], false, bq, (short)0, s[q], false, false);
        }
    }

    // Per-pixel SSIM and tile reduction.
    const float inv_n   = 1.0f / 363.0f;   // N = C*K*K = 3*121
    const float inv_nm1 = 1.0f / 362.0f;   // unbiased variance
    const float C1 = 1.0e-4f;              // 0.01^2
    const float C2 = 9.0e-4f;              // 0.03^2
    float acc = 0.0f;
#pragma unroll
    for (int v = 0; v < 8; ++v) {
        const float sp  = s[0][v], sg  = s[1][v];
        const float spp = s[2][v], sgg = s[3][v], spg = s[4][v];
        const float mup  = sp * inv_n;
        const float mug  = sg * inv_n;
        const float varp = (spp - sp * mup) * inv_nm1;
        const float varg = (sgg - sg * mug) * inv_nm1;
        const float cov  = (spg - sp * mug) * inv_n;
        const float num  = (2.0f * mup * mug + C1) * (2.0f * cov + C2);
        const float den  = (mup * mup + mug * mug + C1) * (varp + varg + C2);
        acc += num / (den + 1.0e-8f);
    }
#pragma unroll
    for (int off = 16; off > 0; off >>= 1)
        acc += __shfl_xor(acc, off, 32);
    if (tid == 0)
        partial[blk] = acc;
}

__global__ __launch_bounds__(256) void
ssim_reduce_kernel(const float* __restrict__ partial, float* __restrict__ out)
{
    __shared__ float sm[256];
    float a = 0.0f;
    for (int i = threadIdx.x; i < NTILES; i += 256)   // fixed order: deterministic
        a += partial[i];
    sm[threadIdx.x] = a;
    __syncthreads();
    for (int sft = 128; sft > 0; sft >>= 1) {
        if (threadIdx.x < sft) sm[threadIdx.x] += sm[threadIdx.x + sft];
        __syncthreads();
    }
    if (threadIdx.x == 0)
        out[0] = 1.0f - sm[0] * (1.0f / (16.0f * 512.0f * 512.0f));
}

extern "C" void kernel_launch(void* const* d_in, const int* in_sizes, int n_in,
                              void* d_out, int out_size, void* d_ws, size_t ws_size,
                              hipStream_t stream)
{
    const float* pred = (const float*)d_in[0];
    const float* gt   = (const float*)d_in[1];
    float* partial = (float*)d_ws;    // NTILES * 4 = 64 KB scratch

    ssim_tile_kernel<<<NTILES, 32, 0, stream>>>(pred, gt, partial);
    ssim_reduce_kernel<<<1, 256, 0, stream>>>(partial, (float*)d_out);
}